// RIMSDecoder_62148176773400
// MI455X (gfx1250) — compile-verified
//
#include <hip/hip_runtime.h>
#include <hip/hip_bf16.h>

// Dims from the reference
#define R_   16
#define NH_  8
#define D_   32
#define CIN  256        // channels
#define RH   128        // R*NH (logit rows)
#define KQ   4096       // R*NH*D
#define BATCH 16
#define XY   1024       // 32*32
#define LSTRIDE 132     // padded LDS row stride (floats) for attn [px][rh]

typedef __attribute__((ext_vector_type(2))) float v2f;
typedef __attribute__((ext_vector_type(8))) float v8f;

// ---------------- prep 1: k,v = per-rim projections ----------------
// k[i] = bk[i] + dot(Wk[i,:], rims[r,:])   (i = r*256 + h*32 + d, flat rh*32+d)
__global__ __launch_bounds__(256) void rims_prep_kv(
    const float* __restrict__ rims, const float* __restrict__ Wk,
    const float* __restrict__ bk,   const float* __restrict__ Wv,
    const float* __restrict__ bv,   float* __restrict__ kflat,
    float* __restrict__ vflat)
{
    int i = blockIdx.x * blockDim.x + threadIdx.x;   // 0..4095
    if (i >= KQ) return;
    int r = i >> 8;                                  // i = r*256 + hd
    const float* rr = rims + r * CIN;
    const float* wk = Wk + (size_t)i * CIN;
    const float* wv = Wv + (size_t)i * CIN;
    float sk = bk[i], sv = bv[i];
    for (int c = 0; c < CIN; ++c) {
        float rc = rr[c];
        sk = fmaf(wk[c], rc, sk);
        sv = fmaf(wv[c], rc, sv);
    }
    kflat[i] = sk;
    vflat[i] = sv;
}

// ---------------- prep 2: fold k into Wq -> A, v into Wm -> M, blog ----------------
// A[rh][c]  = sum_d k[rh*32+d] * Wq[(rh*32+d)*256 + c]
// M[c][rh]  = sum_d Wm[c*4096 + rh*32 + d] * v[rh*32+d]
// blog[rh]  = sum_d k[rh*32+d] * bq[rh*32+d]
__global__ __launch_bounds__(256) void rims_prep_am(
    const float* __restrict__ Wq, const float* __restrict__ bq,
    const float* __restrict__ Wm, const float* __restrict__ kflat,
    const float* __restrict__ vflat, float* __restrict__ Amat,
    float* __restrict__ Mmat, float* __restrict__ blog)
{
    int i = blockIdx.x * blockDim.x + threadIdx.x;
    if (i < RH * CIN) {
        int rh = i >> 8, c = i & 255;
        const float* kp = kflat + rh * D_;
        const float* wp = Wq + (size_t)rh * D_ * CIN + c;
        float s = 0.f;
        for (int d = 0; d < D_; ++d) s = fmaf(kp[d], wp[(size_t)d * CIN], s);
        Amat[rh * CIN + c] = s;
    } else if (i < 2 * RH * CIN) {
        int j = i - RH * CIN;
        int c = j >> 7, rh = j & 127;
        const float* vp = vflat + rh * D_;
        const float* wp = Wm + (size_t)c * KQ + rh * D_;
        float s = 0.f;
        for (int d = 0; d < D_; ++d) s = fmaf(wp[d], vp[d], s);
        Mmat[c * RH + rh] = s;
    } else if (i < 2 * RH * CIN + RH) {
        int rh = i - 2 * RH * CIN;
        const float* kp = kflat + rh * D_;
        const float* bp = bq + rh * D_;
        float s = 0.f;
        for (int d = 0; d < D_; ++d) s = fmaf(kp[d], bp[d], s);
        blog[rh] = s;
    }
}

// ---------------- main fused pass: GEMM1 -> softmax(r) -> GEMM2 ----------------
// One wave per 32-pixel tile. 512 waves total, 2 waves per block.
__global__ __launch_bounds__(64) void rims_main(
    const float* __restrict__ z,    // [B,256,1024]
    const float* __restrict__ Amat, // [128,256]
    const float* __restrict__ Mmat, // [256,128]
    const float* __restrict__ blog, // [128]
    const float* __restrict__ bm,   // [256]
    float* __restrict__ out)        // [B,256,1024]
{
    __shared__ float attn_lds[2][32 * LSTRIDE];   // per-wave attn, [px][rh] padded

    const int lane = threadIdx.x & 31;
    const int wav  = threadIdx.x >> 5;
    const int wave_id = blockIdx.x * 2 + wav;     // 0..511
    const int b   = wave_id >> 5;                 // 32 tiles per image
    const int xy0 = (wave_id & 31) << 5;          // 32 contiguous pixels
    const int col = lane >> 4;                    // wave half (0/1)
    const int lr  = lane & 15;

    const float* zb = z + ((size_t)b * CIN) * XY + xy0;

    // ---- GEMM1: logits[128 x 32] = A @ ztile, accumulated across 8 row-tiles x 2 px-cols
    v8f acc[8][2] = {};
    for (int kk = 0; kk < 64; ++kk) {             // K chunks of 4 over CIN
        const int c0 = kk * 4;
        // B fragment (z): slot(VGPR j, half h) holds row K=c0+j+2h, col N=lane%16
        const float* zp = zb + (size_t)(c0 + 2 * col) * XY + lr;
        v2f bf0, bf1;
        bf0.x = zp[0];        bf0.y = zp[XY];
        bf1.x = zp[16];       bf1.y = zp[XY + 16];
        const float* ap = Amat + c0 + 2 * col + lr * CIN;
#pragma unroll
        for (int mt = 0; mt < 8; ++mt) {
            // A fragment: lane row M=16*mt+lr, K pair (c0+2h, c0+1+2h) -> b64 load
            v2f af = *(const v2f*)(ap + mt * 16 * CIN);
            acc[mt][0] = __builtin_amdgcn_wmma_f32_16x16x4_f32(
                false, af, false, bf0, (short)0, acc[mt][0], false, false);
            acc[mt][1] = __builtin_amdgcn_wmma_f32_16x16x4_f32(
                false, af, false, bf1, (short)0, acc[mt][1], false, false);
        }
    }

    // ---- add logit bias blog[rh]; rh = 16*mt + 8*half + g in D-tile layout
#pragma unroll
    for (int mt = 0; mt < 8; ++mt)
#pragma unroll
        for (int g = 0; g < 8; ++g) {
            float bb = blog[16 * mt + 8 * col + g];
            acc[mt][0][g] += bb;
            acc[mt][1][g] += bb;
        }

    // ---- softmax over rims r (16 values) for fixed head h, pixel px.
    // In-lane: t=0..7 gives r = 2t+col at register index g=h; lane^16 holds the
    // complementary parity for the SAME pixel (px = lane%16).
#pragma unroll
    for (int h = 0; h < 8; ++h) {
#pragma unroll
        for (int ct = 0; ct < 2; ++ct) {
            float m8 = acc[0][ct][h];
#pragma unroll
            for (int t = 1; t < 8; ++t) m8 = fmaxf(m8, acc[t][ct][h]);
            m8 = fmaxf(m8, __shfl_xor(m8, 16, 32));
            float s8 = 0.f;
#pragma unroll
            for (int t = 0; t < 8; ++t) {
                float e = __expf(acc[t][ct][h] - m8);
                acc[t][ct][h] = e;
                s8 += e;
            }
            s8 += __shfl_xor(s8, 16, 32);
            float inv = 1.0f / s8;
#pragma unroll
            for (int t = 0; t < 8; ++t) acc[t][ct][h] *= inv;
        }
    }

    // ---- stage attn into LDS transposed: al[px][rh] (padded) so GEMM2 B-frags are b64 ds loads
    float* al = attn_lds[wav];
#pragma unroll
    for (int t = 0; t < 8; ++t)
#pragma unroll
        for (int g = 0; g < 8; ++g) {
            int rh = 16 * t + 8 * col + g;
            al[lr * LSTRIDE + rh]        = acc[t][0][g];
            al[(16 + lr) * LSTRIDE + rh] = acc[t][1][g];
        }
    __builtin_amdgcn_wave_barrier();   // same-wave DS is in-order; just pin the compiler

    // ---- GEMM2: out[256 x 32] = M[256x128] @ attn[128x32], 16 c-tiles
    const size_t outb = ((size_t)b * CIN) * XY + xy0;
#pragma unroll 1
    for (int ctile = 0; ctile < 16; ++ctile) {
        v8f o0 = {}, o1 = {};
        const float* mp = Mmat + (size_t)(16 * ctile + lr) * RH + 2 * col;
        for (int kk = 0; kk < 32; ++kk) {         // K chunks of 4 over RH=128
            v2f af = *(const v2f*)(mp + kk * 4);
            v2f b0 = *(const v2f*)(al + lr * LSTRIDE + kk * 4 + 2 * col);
            v2f b1 = *(const v2f*)(al + (16 + lr) * LSTRIDE + kk * 4 + 2 * col);
            o0 = __builtin_amdgcn_wmma_f32_16x16x4_f32(
                false, af, false, b0, (short)0, o0, false, false);
            o1 = __builtin_amdgcn_wmma_f32_16x16x4_f32(
                false, af, false, b1, (short)0, o1, false, false);
        }
#pragma unroll
        for (int g = 0; g < 8; ++g) {
            int crow = 16 * ctile + 8 * col + g;
            float bias = bm[crow];
            out[outb + (size_t)crow * XY + lr]      = o0[g] + bias;
            out[outb + (size_t)crow * XY + 16 + lr] = o1[g] + bias;
        }
    }
}

extern "C" void kernel_launch(void* const* d_in, const int* in_sizes, int n_in,
                              void* d_out, int out_size, void* d_ws, size_t ws_size,
                              hipStream_t stream) {
    const float* z    = (const float*)d_in[0];
    const float* rims = (const float*)d_in[1];
    const float* Wk   = (const float*)d_in[2];
    const float* bk   = (const float*)d_in[3];
    const float* Wv   = (const float*)d_in[4];
    const float* bv   = (const float*)d_in[5];
    const float* Wq   = (const float*)d_in[6];
    const float* bq   = (const float*)d_in[7];
    const float* Wm   = (const float*)d_in[8];
    const float* bm   = (const float*)d_in[9];
    float* out = (float*)d_out;

    float* ws    = (float*)d_ws;
    float* kflat = ws;                 // 4096
    float* vflat = ws + 4096;          // 4096
    float* Amat  = ws + 8192;          // 128*256
    float* Mmat  = ws + 40960;         // 256*128
    float* blog  = ws + 73728;         // 128   (total ~295 KB)

    rims_prep_kv<<<dim3(16), dim3(256), 0, stream>>>(rims, Wk, bk, Wv, bv, kflat, vflat);
    rims_prep_am<<<dim3(257), dim3(256), 0, stream>>>(Wq, bq, Wm, kflat, vflat, Amat, Mmat, blog);
    rims_main<<<dim3(256), dim3(64), 0, stream>>>(z, Amat, Mmat, blog, bm, out);
}